// InvConv_23622320128141
// MI455X (gfx1250) — compile-verified
//
#include <hip/hip_runtime.h>
#include <cmath>

typedef float v2f __attribute__((ext_vector_type(2)));
typedef float v8f __attribute__((ext_vector_type(8)));

#define NB    128     // batch
#define IMH   512
#define IMW   512
#define CTX   64
#define HID   128
#define KIN   72      // 8 kernel params + 64 context
#define ROWS  64      // rows per conv block
#define LDSW  520     // padded LDS row (needs >= 514)

// ---------------------------------------------------------------------------
// Hypernet layer 1: h[128,128] = tanh( inp[128,72] @ W1[72,128] + b1 )
// inp[m,k] = (k<8) ? kernel[k] : context[m, k-8]
// One wave computes one 16x16 output tile via V_WMMA_F32_16X16X4_F32 (K=72).
// ---------------------------------------------------------------------------
__device__ __forceinline__ float inp_val(const float* __restrict__ ctx,
                                         const float* __restrict__ kern,
                                         int m, int k) {
  return (k < 8) ? kern[k] : ctx[m * CTX + (k - 8)];
}

__global__ __launch_bounds__(32)
void hyper_layer1(const float* __restrict__ ctx, const float* __restrict__ kern,
                  const float* __restrict__ W1, const float* __restrict__ b1,
                  float* __restrict__ h) {
  const int tn   = blockIdx.x;          // N tile (0..7)
  const int tm   = blockIdx.y;          // M tile (0..7)
  const int lane = threadIdx.x;
  const int half = lane >> 4;           // 0: K=k0,k0+1   1: K=k0+2,k0+3
  const int lrow = lane & 15;
  const int M = tm * 16 + lrow;         // A row for this lane
  const int N = tn * 16 + lrow;         // B/D column for this lane

  v8f c = {};
  for (int k0 = 0; k0 < KIN; k0 += 4) {
    const int ka = k0 + 2 * half;
    v2f a, b;
    a.x = inp_val(ctx, kern, M, ka);
    a.y = inp_val(ctx, kern, M, ka + 1);
    b.x = W1[ka * HID + N];
    b.y = W1[(ka + 1) * HID + N];
    c = __builtin_amdgcn_wmma_f32_16x16x4_f32(false, a, false, b,
                                              (short)0, c, false, false);
  }

  const float bias = b1[N];
  #pragma unroll
  for (int v = 0; v < 8; ++v) {
    const int row = tm * 16 + v + 8 * half;   // D layout: VGPR v, half -> M
    h[row * HID + N] = tanhf(c[v] + bias);
  }
}

// ---------------------------------------------------------------------------
// Hypernet layer 2: w[128,8] = h[128,128] @ W2[128,8] + b2 ; ck[b][0..7]=w,
// ck[b][8]=1.0.  N padded 8->16 with zero B columns; one wave per M tile.
// ---------------------------------------------------------------------------
__global__ __launch_bounds__(32)
void hyper_layer2(const float* __restrict__ h, const float* __restrict__ W2,
                  const float* __restrict__ b2, float* __restrict__ ck9) {
  const int tm   = blockIdx.x;          // M tile (0..7)
  const int lane = threadIdx.x;
  const int half = lane >> 4;
  const int lrow = lane & 15;
  const bool nvalid = (lrow < 8);
  const int  Nc = nvalid ? lrow : 7;    // clamped column -> safe loads

  v8f c = {};
  const float* __restrict__ hrow = h + (tm * 16 + lrow) * HID;
  for (int k0 = 0; k0 < HID; k0 += 4) {
    const int ka = k0 + 2 * half;
    v2f a, b;
    a.x = hrow[ka];
    a.y = hrow[ka + 1];
    const float w0 = W2[ka * 8 + Nc];
    const float w1 = W2[(ka + 1) * 8 + Nc];
    b.x = nvalid ? w0 : 0.0f;
    b.y = nvalid ? w1 : 0.0f;
    c = __builtin_amdgcn_wmma_f32_16x16x4_f32(false, a, false, b,
                                              (short)0, c, false, false);
  }

  if (nvalid) {
    const float bias = b2[lrow];
    #pragma unroll
    for (int v = 0; v < 8; ++v) {
      const int row = tm * 16 + v + 8 * half;
      ck9[row * 9 + lrow] = c[v] + bias;
    }
  }
  if (lrow == 8) {                       // lanes 8 and 24 cover all 16 rows
    #pragma unroll
    for (int v = 0; v < 8; ++v) {
      const int row = tm * 16 + v + 8 * half;
      ck9[row * 9 + 8] = 1.0f;
    }
  }
}

// ---------------------------------------------------------------------------
// UL-padded 3x3 conv: z[b,y,x] = sum_{i,j} ck[b,i,j] * x[b, y+i-2, x+j-2]
// (out-of-range taps are zero; ck[2][2]==1).  Rolling 3-row LDS buffer,
// float4 HBM traffic: each x element read from HBM once (+2 halo rows/chunk).
// ---------------------------------------------------------------------------
__device__ __forceinline__ void load_row(float (*tile)[LDSW],
                                         const float* __restrict__ xb,
                                         int r, int tid) {
  const int s = ((r % 3) + 3) % 3;
  float4 v = make_float4(0.f, 0.f, 0.f, 0.f);
  if (r >= 0)
    v = reinterpret_cast<const float4*>(xb + (size_t)r * IMW)[tid];
  float* dst = &tile[s][2 + 4 * tid];
  dst[0] = v.x; dst[1] = v.y; dst[2] = v.z; dst[3] = v.w;
  if (tid == 0) { tile[s][0] = 0.f; tile[s][1] = 0.f; }   // left pad (x=-2,-1)
}

__global__ __launch_bounds__(128)
void conv_ul3x3(const float* __restrict__ x, const float* __restrict__ ck9,
                float* __restrict__ z, float* __restrict__ ladj) {
  __shared__ float tile[3][LDSW];
  const int b     = blockIdx.y;
  const int chunk = blockIdx.x;
  const int tid   = threadIdx.x;
  const int y0    = chunk * ROWS;

  const float* __restrict__ ckb = ck9 + b * 9;
  const float c00 = ckb[0], c01 = ckb[1], c02 = ckb[2];
  const float c10 = ckb[3], c11 = ckb[4], c12 = ckb[5];
  const float c20 = ckb[6], c21 = ckb[7], c22 = ckb[8];

  const float* __restrict__ xb = x + (size_t)b * IMH * IMW;
  float* __restrict__       zb = z + (size_t)b * IMH * IMW;

  if (chunk == 0 && tid == 0) ladj[b] = 0.0f;   // second tuple output = zeros

  load_row(tile, xb, y0 - 2, tid);              // halo rows (zeros if y<0)
  load_row(tile, xb, y0 - 1, tid);

  for (int y = y0; y < y0 + ROWS; ++y) {
    load_row(tile, xb, y, tid);
    const int yn = (y + 1 < IMH) ? (y + 1) : y;
    __builtin_prefetch(xb + (size_t)yn * IMW + 4 * tid, 0, 0);
    __syncthreads();

    // slot(y-2) == (y+1)%3, slot(y-1) == (y+2)%3, slot(y) == y%3
    const float* __restrict__ r0 = tile[(y + 1) % 3];
    const float* __restrict__ r1 = tile[(y + 2) % 3];
    const float* __restrict__ r2 = tile[y % 3];
    const int xo = 4 * tid;   // LDS idx p holds x[p-2]; tap j reads idx xo+q+j

    float o[4];
    #pragma unroll
    for (int q = 0; q < 4; ++q) {
      const int p = xo + q;
      float acc =      c00 * r0[p];
      acc = fmaf(c01, r0[p + 1], acc);
      acc = fmaf(c02, r0[p + 2], acc);
      acc = fmaf(c10, r1[p],     acc);
      acc = fmaf(c11, r1[p + 1], acc);
      acc = fmaf(c12, r1[p + 2], acc);
      acc = fmaf(c20, r2[p],     acc);
      acc = fmaf(c21, r2[p + 1], acc);
      acc = fmaf(c22, r2[p + 2], acc);
      o[q] = acc;
    }
    reinterpret_cast<float4*>(zb + (size_t)y * IMW)[tid] =
        make_float4(o[0], o[1], o[2], o[3]);

    __syncthreads();   // slot(y-2) gets overwritten next iteration
  }
}

// ---------------------------------------------------------------------------
extern "C" void kernel_launch(void* const* d_in, const int* in_sizes, int n_in,
                              void* d_out, int out_size, void* d_ws, size_t ws_size,
                              hipStream_t stream) {
  (void)in_sizes; (void)n_in; (void)out_size; (void)ws_size;
  const float* x    = (const float*)d_in[0];
  const float* ctx  = (const float*)d_in[1];
  const float* kern = (const float*)d_in[2];
  const float* W1   = (const float*)d_in[3];
  const float* b1   = (const float*)d_in[4];
  const float* W2   = (const float*)d_in[5];
  const float* b2   = (const float*)d_in[6];

  float* z    = (float*)d_out;                      // [128,1,512,512]
  float* ladj = z + (size_t)NB * IMH * IMW;         // [128]

  float* h   = (float*)d_ws;                        // [128,128] f32 (64 KB)
  float* ck9 = h + NB * HID;                        // [128,9]  f32

  hipLaunchKernelGGL(hyper_layer1, dim3(HID / 16, NB / 16), dim3(32), 0, stream,
                     ctx, kern, W1, b1, h);
  hipLaunchKernelGGL(hyper_layer2, dim3(NB / 16), dim3(32), 0, stream,
                     h, W2, b2, ck9);
  hipLaunchKernelGGL(conv_ul3x3, dim3(IMH / ROWS, NB), dim3(128), 0, stream,
                     x, ck9, z, ladj);
}